// GridSamplePScan_63359357551173
// MI455X (gfx1250) — compile-verified
//
#include <hip/hip_runtime.h>
#include <stdint.h>

// Problem constants (B,C,H,W) = (4,32,512,512), fp32.
#define WIDTH    512
#define HEIGHT   512
#define CHANNELS 32
#define HWSZ     (WIDTH * HEIGHT)       // 262144
#define CHW      (CHANNELS * HWSZ)      // 8388608

typedef float fvec4 __attribute__((ext_vector_type(4)));   // native vector for b128 + NT

#ifndef __has_builtin
#define __has_builtin(x) 0
#endif

#if __has_builtin(__builtin_amdgcn_global_load_async_to_lds_b128)
#define USE_ASYNC_LDS 1
#else
#define USE_ASYNC_LDS 0
#endif

#if USE_ASYNC_LDS
typedef int async_v4i __attribute__((ext_vector_type(4)));
typedef __attribute__((address_space(1))) async_v4i* g_v4i_ptr;   // global
typedef __attribute__((address_space(3))) async_v4i* l_v4i_ptr;   // LDS

// cpol = 1 -> TH=NT: residual is touch-once; don't let it evict x from L2.
__device__ __forceinline__ void async_copy_b128_nt(const float* gsrc, float* ldst) {
  __builtin_amdgcn_global_load_async_to_lds_b128(
      (g_v4i_ptr)(void*)const_cast<float*>(gsrc),
      (l_v4i_ptr)(void*)ldst, /*offset=*/0, /*cpol=*/1);
}
#endif

template <int N>
__device__ __forceinline__ void wait_asynccnt() {
#if __has_builtin(__builtin_amdgcn_s_wait_asynccnt)
  __builtin_amdgcn_s_wait_asynccnt(N);
#else
  asm volatile("s_wait_asynccnt %0" ::"n"(N) : "memory");
#endif
}

// One thread = 4 consecutive pixels of one (b, h) row; loops over 32 channels.
// Block = 256 threads = 1024 pixels = exactly 2 rows. Grid = 1024 blocks.
__global__ __launch_bounds__(256)
void warp_bilinear_add(const float* __restrict__ x,
                       const float* __restrict__ flow,
                       const float* __restrict__ residual,
                       float* __restrict__ out) {
#if USE_ASYNC_LDS
  __shared__ float lds_res[2][256 * 4];   // 8 KB double buffer for residual
#endif
  const int tid = threadIdx.x;
  const int p4  = blockIdx.x * 256 + tid;     // pixel-quad index
  const int w4  = p4 & (WIDTH / 4 - 1);       // 128 quads per row
  const int row = p4 >> 7;                    // b*H + h
  const int h   = row & (HEIGHT - 1);
  const int b   = row >> 9;
  const int w   = w4 * 4;

  const long rowoff = (long)h * WIDTH + w;

  // flow[b,0,h,w..w+3] and flow[b,1,h,w..w+3]: touch-once streams -> NT loads.
  const float* flow_b = flow + (long)b * 2 * HWSZ;
  const fvec4 fx4 = __builtin_nontemporal_load((const fvec4*)(flow_b + rowoff));
  const fvec4 fy4 = __builtin_nontemporal_load((const fvec4*)(flow_b + HWSZ + rowoff));
  const float fxa[4] = {fx4.x, fx4.y, fx4.z, fx4.w};
  const float fya[4] = {fy4.x, fy4.y, fy4.z, fy4.w};

  const float gy = ((float)h + 0.5f) * (2.0f / HEIGHT) - 1.0f;

  // Per-pixel interpolation state, computed once, reused for all 32 channels.
  int   itl[4], itr[4], ibl[4], ibr[4];
  float wtl[4], wtr[4], wbl[4], wbr[4];

#pragma unroll
  for (int j = 0; j < 4; ++j) {
    const float gx = ((float)(w + j) + 0.5f) * (2.0f / WIDTH) - 1.0f;
    // circular wrap in x: t = mod(grid_x + 1, 2) in [0,2);  real_x = t*W/2 - 0.5
    float t = gx + fxa[j] + 1.0f;
    t = t - floorf(t * 0.5f) * 2.0f;
    const float real_x = t * (WIDTH * 0.5f) - 0.5f;
    const float real_y = (gy + fya[j] + 1.0f) * (HEIGHT * 0.5f) - 0.5f;

    const float x0 = floorf(real_x);
    const float y0 = floorf(real_y);
    const float dx = real_x - x0;
    const float dy = real_y - y0;
    const int ix0 = (int)x0, iy0 = (int)y0;
    const int ix1 = ix0 + 1, iy1 = iy0 + 1;

    const bool vx0 = (unsigned)ix0 < (unsigned)WIDTH;
    const bool vx1 = (unsigned)ix1 < (unsigned)WIDTH;
    const bool vy0 = (unsigned)iy0 < (unsigned)HEIGHT;
    const bool vy1 = (unsigned)iy1 < (unsigned)HEIGHT;

    const int cx0 = min(max(ix0, 0), WIDTH - 1);
    const int cx1 = min(max(ix1, 0), WIDTH - 1);
    const int cy0 = min(max(iy0, 0), HEIGHT - 1);
    const int cy1 = min(max(iy1, 0), HEIGHT - 1);

    itl[j] = cy0 * WIDTH + cx0;
    itr[j] = cy0 * WIDTH + cx1;
    ibl[j] = cy1 * WIDTH + cx0;
    ibr[j] = cy1 * WIDTH + cx1;

    // OOB taps contribute 0: fold validity into the weight (value*0 == 0).
    wtl[j] = (vx0 && vy0) ? (1.0f - dx) * (1.0f - dy) : 0.0f;
    wtr[j] = (vx1 && vy0) ? dx * (1.0f - dy)          : 0.0f;
    wbl[j] = (vx0 && vy1) ? (1.0f - dx) * dy          : 0.0f;
    wbr[j] = (vx1 && vy1) ? dx * dy                   : 0.0f;
  }

  const long  base_b = (long)b * CHW;
  const float* xb = x + base_b;             // x gathers: default RT policy (L2-resident)
  const float* rb = residual + base_b + rowoff;
  float*       ob = out + base_b + rowoff;

#if USE_ASYNC_LDS
  async_copy_b128_nt(rb, &lds_res[0][tid * 4]);   // preload channel 0 residual
#endif

#pragma unroll 2
  for (int c = 0; c < CHANNELS; ++c) {
    const float* xc = xb + (long)c * HWSZ;

#if USE_ASYNC_LDS
    if (c + 1 < CHANNELS) {   // prefetch next channel's residual into LDS
      async_copy_b128_nt(rb + (long)(c + 1) * HWSZ, &lds_res[(c + 1) & 1][tid * 4]);
    }
#endif

    // 16 gathers per thread (4 taps x 4 pixels), all L2-resident.
    float acc[4];
#pragma unroll
    for (int j = 0; j < 4; ++j) {
      acc[j] = wtl[j] * xc[itl[j]] + wtr[j] * xc[itr[j]] +
               wbl[j] * xc[ibl[j]] + wbr[j] * xc[ibr[j]];
    }

    fvec4 res;
#if USE_ASYNC_LDS
    if (c + 1 < CHANNELS) wait_asynccnt<1>();   // current stage done, next in flight
    else                  wait_asynccnt<0>();
    asm volatile("" ::: "memory");              // fence the LDS read past the wait
    res = *(const fvec4*)&lds_res[c & 1][tid * 4];
#else
    res = __builtin_nontemporal_load((const fvec4*)(rb + (long)c * HWSZ));
#endif

    fvec4 o;
    o.x = acc[0] + res.x;
    o.y = acc[1] + res.y;
    o.z = acc[2] + res.z;
    o.w = acc[3] + res.w;
    // out is write-once: NT store keeps it from displacing x in L2.
    __builtin_nontemporal_store(o, (fvec4*)(ob + (long)c * HWSZ));
  }
}

extern "C" void kernel_launch(void* const* d_in, const int* in_sizes, int n_in,
                              void* d_out, int out_size, void* d_ws, size_t ws_size,
                              hipStream_t stream) {
  const float* x        = (const float*)d_in[0];
  const float* flow     = (const float*)d_in[1];
  const float* residual = (const float*)d_in[2];
  float* out            = (float*)d_out;

  // B*H*W/4 threads = 262144 -> 1024 blocks of 256 (8 wave32 per block)
  dim3 grid(1024), block(256);
  warp_bilinear_add<<<grid, block, 0, stream>>>(x, flow, residual, out);
}